// FeatureInterpolator_67078799229396
// MI455X (gfx1250) — compile-verified
//
#include <hip/hip_runtime.h>
#include <hip/hip_bf16.h>

typedef __attribute__((ext_vector_type(2))) float v2f;
typedef __attribute__((ext_vector_type(8))) float v8f;

#define B_   8
#define N1_  8192
#define N2_  2048
#define C1_  256
#define C2_  512
#define KNN  3
#define EPS_ 1e-10f
#define OFS_ 0.015625f   // 2^-6: keeps WMMA scores strictly positive

__device__ __forceinline__ unsigned umin_(unsigned a, unsigned b) { return a < b ? a : b; }
__device__ __forceinline__ unsigned umax_(unsigned a, unsigned b) { return a > b ? a : b; }

// Insert packed candidate into sorted triple u0<=u1<=u2 (unsigned compare ==
// float compare for positive floats). 5 min/max_u32 ops, no canonicalize.
__device__ __forceinline__ void insertp(unsigned p, unsigned& u0, unsigned& u1, unsigned& u2) {
  unsigned m0 = umax_(p, u0);
  u0 = umin_(p, u0);
  unsigned m1 = umax_(m0, u1);   // max3(p, u0, u1)
  u1 = umin_(m0, u1);
  u2 = umin_(m1, u2);
}

// Kernel 1: 3-NN search. Block = 256 threads (8 waves), each wave owns 16
// queries; block covers 128 queries. Keys for this batch live in LDS,
// pre-transformed to (-2x,-2y,-2z,||k||^2). Scores via V_WMMA_F32_16X16X4_F32:
//   A row m = [qx,qy,qz,1],  B col n = [-2kx,-2ky,-2kz,||k||^2],
//   C[m] = ||q_m||^2 + 2^-6   (preloaded accumulator, hoisted out of the loop)
// => D(m,n) = ||q-k||^2 + 2^-6  (positive => uint ordering is float ordering).
// Candidates packed as (score_bits & ~0x7FF) | key_index; top-3 kept with a
// 5-op unsigned min/max sorting network. B tiles are software-pipelined so the
// ds_load latency overlaps the previous tile's insert network. Exact distances
// are recomputed from the selected indices for the weights.
__global__ __launch_bounds__(256)
void knn_kernel(const float* __restrict__ qxyz, const float* __restrict__ kxyz,
                int* __restrict__ out_idx, float* __restrict__ out_w) {
  __shared__ float4 skey[N2_];
  __shared__ float  sqq[128];   // also serves as safe overrun target for the
                                // one-tile B prefetch past skey (value unused)

  const int b   = blockIdx.y;
  const int qb0 = blockIdx.x * 128;
  const int tid = threadIdx.x;

  const float* kx = kxyz + (size_t)b * 3 * N2_;
  const float* ky = kx + N2_;
  const float* kz = ky + N2_;
  for (int k = tid; k < N2_; k += 256) {
    float x = kx[k], y = ky[k], z = kz[k];
    skey[k] = make_float4(-2.0f * x, -2.0f * y, -2.0f * z, x * x + y * y + z * z);
  }

  const float* qx = qxyz + (size_t)b * 3 * N1_;
  const float* qy = qx + N1_;
  const float* qz = qy + N1_;
  if (tid < 128) {
    int q = qb0 + tid;
    float x = qx[q], y = qy[q], z = qz[q];
    sqq[tid] = x * x + y * y + z * z;
  }
  __syncthreads();

  const int wave  = tid >> 5;
  const int lane  = tid & 31;
  const int lo    = lane & 15;
  const int hi    = lane >> 4;        // 0: K=0,1 half; 1: K=2,3 half
  const int qbase = qb0 + wave * 16;

  // A (16x4 fp32, WMMA layout): lane m in [0,16): {qx, qy}; lane m+16: {qz, 1}
  int qi = qbase + lo;
  float ax = qx[qi], ay = qy[qi], az = qz[qi];
  v2f A;
  A[0] = hi ? az : ax;
  A[1] = hi ? 1.0f : ay;

  // Accumulator preload: C[r] = ||q_row||^2 + eps for row = 8*hi + r
  // (uniform across the 16 lanes of each column group -> LDS broadcast read).
  v8f cinit;
#pragma unroll
  for (int r = 0; r < 8; ++r)
    cinit[r] = sqq[wave * 16 + hi * 8 + r] + OFS_;

  unsigned u0[8], u1[8], u2[8];
#pragma unroll
  for (int r = 0; r < 8; ++r)
    u0[r] = u1[r] = u2[r] = 0x7F000000u;  // large positive float bits, idx 0

  // B (4x16 fp32, WMMA layout): lane n: {-2kx,-2ky}; lane n+16: {-2kz, kk}
  const float2* bbase = (const float2*)&skey[lo] + hi;
  float2 cur = bbase[0];

#pragma unroll 2
  for (int t = 0; t < N2_ / 16; ++t) {
    float2 nxt = bbase[(t + 1) * 32];   // prefetch next tile (last one lands
                                        // in sqq space: in-bounds, unused)
    v2f Bv; Bv[0] = cur.x; Bv[1] = cur.y;
    v8f acc = __builtin_amdgcn_wmma_f32_16x16x4_f32(
        /*neg_a=*/false, A, /*neg_b=*/false, Bv,
        /*c_mod=*/(short)0, cinit, /*reuse_a=*/false, /*reuse_b=*/false);
    unsigned key = (unsigned)(t * 16 + lo);
    // C/D layout: lane holds column n=key for rows m = 8*hi + r
#pragma unroll
    for (int r = 0; r < 8; ++r) {
      unsigned p = (__float_as_uint(acc[r]) & 0xFFFFF800u) | key;
      insertp(p, u0[r], u1[r], u2[r]);
    }
    cur = nxt;
  }

  // Butterfly merge across the 16 lanes holding the same 8 rows.
#pragma unroll
  for (int m = 1; m <= 8; m <<= 1) {
#pragma unroll
    for (int r = 0; r < 8; ++r) {
      unsigned p0 = (unsigned)__shfl_xor((int)u0[r], m);
      unsigned p1 = (unsigned)__shfl_xor((int)u1[r], m);
      unsigned p2 = (unsigned)__shfl_xor((int)u2[r], m);
      insertp(p0, u0[r], u1[r], u2[r]);
      insertp(p1, u0[r], u1[r], u2[r]);
      insertp(p2, u0[r], u1[r], u2[r]);
    }
  }

  if (lo == 0) {  // lane 0 writes rows 0..7, lane 16 writes rows 8..15
#pragma unroll
    for (int r = 0; r < 8; ++r) {
      int row = hi * 8 + r;
      int q   = qbase + row;
      float qxv = qx[q], qyv = qy[q], qzv = qz[q];
      float qq  = sqq[wave * 16 + row];

      int j0 = (int)(u0[r] & 0x7FFu);
      int j1 = (int)(u1[r] & 0x7FFu);
      int j2 = (int)(u2[r] & 0x7FFu);

      // Recompute exact squared distances from LDS key data.
      float4 k0 = skey[j0], k1 = skey[j1], k2 = skey[j2];
      float e0 = fmaxf(qq + k0.w + qxv * k0.x + qyv * k0.y + qzv * k0.z, EPS_);
      float e1 = fmaxf(qq + k1.w + qxv * k1.x + qyv * k1.y + qzv * k1.z, EPS_);
      float e2 = fmaxf(qq + k2.w + qxv * k2.x + qyv * k2.y + qzv * k2.z, EPS_);

      float v0 = 1.0f / e0, v1 = 1.0f / e1, v2 = 1.0f / e2;
      float inv = 1.0f / (v0 + v1 + v2);
      size_t base = ((size_t)b * N1_ + q) * KNN;
      out_idx[base + 0] = j0;
      out_idx[base + 1] = j1;
      out_idx[base + 2] = j2;
      out_w[base + 0] = v0 * inv;
      out_w[base + 1] = v1 * inv;
      out_w[base + 2] = v2 * inv;
    }
  }
}

// Kernel 2: weighted gather of key features + concat of query features.
// Block per (batch, 64-query tile); idx/weights staged in LDS. Each thread
// owns a float4 along N1 -> b128 loads/stores saturate HBM (201 MB output);
// the 3-way gathers hit the L2-resident 4 MB key_feature slice.
#define TILE_N 64
__global__ __launch_bounds__(256)
void interp_kernel(const float* __restrict__ qf, const float* __restrict__ kf,
                   const int* __restrict__ idx, const float* __restrict__ w,
                   float* __restrict__ out) {
  __shared__ int   sidx[TILE_N * KNN];
  __shared__ float sw[TILE_N * KNN];

  const int b  = blockIdx.y;
  const int n0 = blockIdx.x * TILE_N;
  const int tid = threadIdx.x;

  for (int j = tid; j < TILE_N * KNN; j += 256) {
    size_t base = ((size_t)b * N1_ + n0) * KNN + j;
    sidx[j] = idx[base];
    sw[j]   = w[base];
  }
  __syncthreads();

  const int nv = tid & 15;   // which float4 of the 64 queries
  const int cg = tid >> 4;   // 16 channel groups
  const int n  = n0 + nv * 4;

  int   j0[4], j1[4], j2[4];
  float w0[4], w1[4], w2[4];
#pragma unroll
  for (int e = 0; e < 4; ++e) {
    int nl = nv * 4 + e;
    j0[e] = sidx[nl * KNN + 0];
    j1[e] = sidx[nl * KNN + 1];
    j2[e] = sidx[nl * KNN + 2];
    w0[e] = sw[nl * KNN + 0];
    w1[e] = sw[nl * KNN + 1];
    w2[e] = sw[nl * KNN + 2];
  }

  const float* kfb  = kf + (size_t)b * C2_ * N2_;
  const float* qfb  = qf + (size_t)b * C1_ * N1_;
  float*       outb = out + (size_t)b * (C1_ + C2_) * N1_;

  for (int c = cg; c < C2_; c += 16) {
    const float* row = kfb + (size_t)c * N2_;
    float4 v;
    v.x = w0[0] * row[j0[0]] + w1[0] * row[j1[0]] + w2[0] * row[j2[0]];
    v.y = w0[1] * row[j0[1]] + w1[1] * row[j1[1]] + w2[1] * row[j2[1]];
    v.z = w0[2] * row[j0[2]] + w1[2] * row[j1[2]] + w2[2] * row[j2[2]];
    v.w = w0[3] * row[j0[3]] + w1[3] * row[j1[3]] + w2[3] * row[j2[3]];
    *(float4*)(outb + (size_t)c * N1_ + n) = v;
  }
  for (int c = cg; c < C1_; c += 16) {
    float4 v = *(const float4*)(qfb + (size_t)c * N1_ + n);
    *(float4*)(outb + (size_t)(C2_ + c) * N1_ + n) = v;
  }
}

extern "C" void kernel_launch(void* const* d_in, const int* in_sizes, int n_in,
                              void* d_out, int out_size, void* d_ws, size_t ws_size,
                              hipStream_t stream) {
  const float* qxyz = (const float*)d_in[0];
  const float* kxyz = (const float*)d_in[1];
  const float* qf   = (const float*)d_in[2];
  const float* kf   = (const float*)d_in[3];
  float* out = (float*)d_out;

  // workspace: indices then weights
  int*   ws_idx = (int*)d_ws;
  float* ws_w   = (float*)((int*)d_ws + (size_t)B_ * N1_ * KNN);

  dim3 g1(N1_ / 128, B_);
  knn_kernel<<<g1, 256, 0, stream>>>(qxyz, kxyz, ws_idx, ws_w);

  dim3 g2(N1_ / TILE_N, B_);
  interp_kernel<<<g2, 256, 0, stream>>>(qf, kf, ws_idx, ws_w, out);
}